// AGIFDecoder_22797686408108
// MI455X (gfx1250) — compile-verified
//
#include <hip/hip_runtime.h>
#include <hip/hip_bf16.h>
#include <math.h>

// Sizes
#define B_   32
#define S_   64
#define DIN_ 256
#define H_   256
#define E_   128
#define V_   128
#define NN_  128      // graph nodes (NI+1)
#define GH_  64
#define NH_  4
#define DX_  384      // DIN + E
#define G4H_ 1024     // 4*H
#define M_   2048     // B*S

typedef float v2f __attribute__((ext_vector_type(2)));
typedef float v8f __attribute__((ext_vector_type(8)));

// fp32 WMMA: D[16x16] += A[16x4] * B[4x16]
// A layout: lanes 0-15 -> rows, vgpr{0,1}=K{0,1}; lanes 16-31 -> K{2,3}
// B layout: lanes 0-15 -> cols, vgpr{0,1}=K{0,1}; lanes 16-31 -> K{2,3}
// D layout: vgpr r -> row r (lanes 0-15) / row r+8 (lanes 16-31), col = lane&15
__device__ __forceinline__ v8f wmma4(v2f a, v2f b, v8f c) {
  return __builtin_amdgcn_wmma_f32_16x16x4_f32(false, a, false, b, (short)0, c,
                                               false, false);
}

__device__ __forceinline__ float leakyf(float x) { return x >= 0.f ? x : 0.2f * x; }
__device__ __forceinline__ float eluf(float x)   { return x > 0.f ? x : expf(x) - 1.f; }
__device__ __forceinline__ float sigf(float x)   { return 1.f / (1.f + expf(-x)); }

// ---------------------------------------------------------------- utilities
__global__ void k_transpose(const float* __restrict__ src, float* __restrict__ dst,
                            int R, int C) {
  int idx = blockIdx.x * blockDim.x + threadIdx.x;
  if (idx < R * C) {
    int r = idx / C, c = idx % C;
    dst[c * R + r] = src[idx];
  }
}

// Build LSTM input X[2048][384] = [hidden | shifted teacher-forced embedding]
__global__ void k_build_x(const float* __restrict__ hidden,
                          const int* __restrict__ forced,
                          const float* __restrict__ emb,
                          const float* __restrict__ init, float* __restrict__ X) {
  int idx = blockIdx.x * blockDim.x + threadIdx.x;  // exact grid: 2048*384
  int m = idx / DX_, k = idx % DX_;
  int b = m >> 6, s = m & 63;
  float v;
  if (k < DIN_) {
    v = hidden[m * DIN_ + k];
  } else {
    int e = k - DIN_;
    v = (s == 0) ? init[e] : emb[forced[b * S_ + s - 1] * E_ + e];
  }
  X[idx] = v;
}

// Intent-node projections per head: HINT[k][j][g] = intent[j-1] . Wh[k][:,g]
// plus f1_int/f2_int = HINT . ah halves. Row j==0 zeroed.
__global__ void k_hint(const float* __restrict__ intent, const float* __restrict__ Wh,
                       const float* __restrict__ ah, float* __restrict__ HINT,
                       float* __restrict__ F1I, float* __restrict__ F2I) {
  int k = blockIdx.x, j = blockIdx.y, g = threadIdx.x;  // grid (4,128), 64 thr
  __shared__ float xe[H_];
  __shared__ float red[GH_];
  if (j == 0) {
    if (g == 0) { F1I[k * NN_] = 0.f; F2I[k * NN_] = 0.f; }
    HINT[(k * NN_) * GH_ + g] = 0.f;
    return;
  }
  for (int f = g; f < H_; f += GH_) xe[f] = intent[(j - 1) * H_ + f];
  __syncthreads();
  float acc = 0.f;
  const float* wcol = Wh + (k * H_) * GH_ + g;  // Wh[k][f][g], stride GH_
  for (int f = 0; f < H_; ++f) acc += xe[f] * wcol[f * GH_];
  HINT[(k * NN_ + j) * GH_ + g] = acc;
  red[g] = acc * ah[k * 2 * GH_ + g];
  __syncthreads();
  for (int off = 32; off; off >>= 1) { if (g < off) red[g] += red[g + off]; __syncthreads(); }
  if (g == 0) F1I[k * NN_ + j] = red[0];
  __syncthreads();
  red[g] = acc * ah[k * 2 * GH_ + GH_ + g];
  __syncthreads();
  for (int off = 32; off; off >>= 1) { if (g < off) red[g] += red[g + off]; __syncthreads(); }
  if (g == 0) F2I[k * NN_ + j] = red[0];
}

// WOA[sel][f] = sum_n Wo[f][n] * ao[sel*256+n]   (folds layer-2 attn vectors)
__global__ void k_woa(const float* __restrict__ Wo, const float* __restrict__ ao,
                      float* __restrict__ WOA) {
  int sel = blockIdx.x, f = threadIdx.x;
  const float* a = ao + sel * H_;
  const float* row = Wo + f * H_;
  float acc = 0.f;
  for (int n = 0; n < H_; ++n) acc += row[n] * a[n];
  WOA[sel * H_ + f] = acc;
}

// ---------------------------------------------------- GEMM1: P = X@WihT + bias
__global__ void k_gemm_p(const float* __restrict__ X, const float* __restrict__ WihT,
                         const float* __restrict__ b_ih, const float* __restrict__ b_hh,
                         float* __restrict__ P) {
  int wave = threadIdx.x >> 5, lane = threadIdx.x & 31;
  int tile = blockIdx.x * 8 + wave;  // 8192 tiles: 128 x 64
  int m0 = (tile >> 6) * 16, n0 = (tile & 63) * 16;
  int row = m0 + (lane & 15), col = n0 + (lane & 15);
  int kh = (lane >> 4) << 1;
  v8f acc = {};
  for (int k0 = 0; k0 < DX_; k0 += 4) {
    v2f a, b;
    a.x = X[row * DX_ + k0 + kh];
    a.y = X[row * DX_ + k0 + kh + 1];
    b.x = WihT[(k0 + kh) * G4H_ + col];
    b.y = WihT[(k0 + kh + 1) * G4H_ + col];
    acc = wmma4(a, b, acc);
  }
  float bias = b_ih[col] + b_hh[col];
  int rbase = m0 + ((lane >> 4) << 3);
  for (int r = 0; r < 8; ++r) P[(rbase + r) * G4H_ + col] = acc[r] + bias;
}

// ------------------------------------------------- LSTM: persistent one-WGP
// LDS: hcur[32][260] (padded), gbuf[32][1024]
#define HCS 260
__global__ void __launch_bounds__(1024) k_lstm(const float* __restrict__ P,
                                               const float* __restrict__ WhhT,
                                               float* __restrict__ LO) {
  extern __shared__ float sm[];
  float* hcur = sm;                 // 32*HCS
  float* gbuf = sm + B_ * HCS;      // 32*1024
  int tid = threadIdx.x;
  int wave = tid >> 5, lane = tid & 31;
  for (int i = tid; i < B_ * HCS; i += 1024) hcur[i] = 0.f;
  float creg[8];
#pragma unroll
  for (int p = 0; p < 8; ++p) creg[p] = 0.f;
  __syncthreads();

  for (int t = 0; t < S_; ++t) {
    // recurrent GEMM: [32,256] @ [256,1024] -> gbuf; 128 tiles / 32 waves
    for (int tt = 0; tt < 4; ++tt) {
      int tile = wave * 4 + tt;
      int m0 = (tile >> 6) * 16, n0 = (tile & 63) * 16;
      int row = m0 + (lane & 15), col = n0 + (lane & 15);
      int kh = (lane >> 4) << 1;
      v8f acc = {};
      for (int k0 = 0; k0 < H_; k0 += 4) {
        v2f a, b;
        a.x = hcur[row * HCS + k0 + kh];
        a.y = hcur[row * HCS + k0 + kh + 1];
        b.x = WhhT[(k0 + kh) * G4H_ + col];
        b.y = WhhT[(k0 + kh + 1) * G4H_ + col];
        acc = wmma4(a, b, acc);
      }
      int rbase = m0 + ((lane >> 4) << 3);
#pragma unroll
      for (int r = 0; r < 8; ++r) gbuf[(rbase + r) * G4H_ + col] = acc[r];
    }
    __syncthreads();
#pragma unroll
    for (int p = 0; p < 8; ++p) {
      int idx = tid * 8 + p;          // 0..8191 -> (b,u)
      int b = idx >> 8, u = idx & 255;
      const float* Pg = P + (b * S_ + t) * G4H_;
      const float* Gg = gbuf + b * G4H_;
      float gi = Pg[u] + Gg[u];
      float gf = Pg[H_ + u] + Gg[H_ + u];
      float gg = Pg[2 * H_ + u] + Gg[2 * H_ + u];
      float go = Pg[3 * H_ + u] + Gg[3 * H_ + u];
      float cn = sigf(gf) * creg[p] + sigf(gi) * tanhf(gg);
      creg[p] = cn;
      float h = sigf(go) * tanhf(cn);
      hcur[b * HCS + u] = h;
      LO[(b * S_ + t) * H_ + u] = h;
    }
    __syncthreads();
  }
}

// --------------------------------- HT = LO @ [Wh heads col-concat]  [2048,256]
__global__ void k_gemm_ht(const float* __restrict__ LO, const float* __restrict__ Wh,
                          float* __restrict__ HT) {
  int wave = threadIdx.x >> 5, lane = threadIdx.x & 31;
  int tile = blockIdx.x * 8 + wave;  // 2048 tiles: 128 x 16
  int m0 = (tile >> 4) * 16, n0 = (tile & 15) * 16;
  int row = m0 + (lane & 15), col = n0 + (lane & 15);
  int head = col >> 6, g = col & 63;
  const float* wb = Wh + (head * H_) * GH_ + g;  // Wh[head][f][g], stride GH_
  int kh = (lane >> 4) << 1;
  v8f acc = {};
  for (int k0 = 0; k0 < H_; k0 += 4) {
    v2f a, b;
    a.x = LO[row * H_ + k0 + kh];
    a.y = LO[row * H_ + k0 + kh + 1];
    b.x = wb[(k0 + kh) * GH_];
    b.y = wb[(k0 + kh + 1) * GH_];
    acc = wmma4(a, b, acc);
  }
  int rbase = m0 + ((lane >> 4) << 3);
  for (int r = 0; r < 8; ++r) HT[(rbase + r) * H_ + col] = acc[r];
}

// ----------------- per (head,batch): masked-exp matrix, Z1 rowsums, V1 = W@h
#define WMS 129
__global__ void k_g1(const int* __restrict__ adj, const float* __restrict__ HINT,
                     const float* __restrict__ F1I, const float* __restrict__ F2I,
                     float* __restrict__ Z1, float* __restrict__ V1) {
  extern __shared__ float sm[];
  float* Wm = sm;                  // [128][WMS]
  float* hs = sm + NN_ * WMS;      // [128][64]
  int k = blockIdx.x, b = blockIdx.y;
  int tid = threadIdx.x;
  for (int i = tid; i < NN_ * GH_; i += 256) hs[i] = HINT[k * NN_ * GH_ + i];
  for (int idx = tid; idx < NN_ * NN_; idx += 256) {
    int i = idx >> 7, j = idx & 127;
    float w = 0.f;
    if (i > 0 && j > 0 && adj[(b * NN_ + i) * NN_ + j] > 0)
      w = expf(leakyf(F1I[k * NN_ + i] + F2I[k * NN_ + j]));
    Wm[i * WMS + j] = w;
  }
  __syncthreads();
  if (tid < NN_) {
    float s = 0.f;
    for (int j = 0; j < NN_; ++j) s += Wm[tid * WMS + j];
    Z1[(k * B_ + b) * NN_ + tid] = s;
  }
  __syncthreads();
  // V1 = Wm[128x128] @ hs[128x64]; 32 tiles / 8 waves
  int wave = tid >> 5, lane = tid & 31;
  for (int tt = 0; tt < 4; ++tt) {
    int tile = wave * 4 + tt;
    int m0 = (tile >> 2) * 16, n0 = (tile & 3) * 16;
    int row = m0 + (lane & 15), col = n0 + (lane & 15);
    int kh = (lane >> 4) << 1;
    v8f acc = {};
    for (int k0 = 0; k0 < NN_; k0 += 4) {
      v2f a, bb;
      a.x = Wm[row * WMS + k0 + kh];
      a.y = Wm[row * WMS + k0 + kh + 1];
      bb.x = hs[(k0 + kh) * GH_ + col];
      bb.y = hs[(k0 + kh + 1) * GH_ + col];
      acc = wmma4(a, bb, acc);
    }
    int rbase = m0 + ((lane >> 4) << 3);
    for (int r = 0; r < 8; ++r)
      V1[((k * B_ + b) * NN_ + rbase + r) * GH_ + col] = acc[r];
  }
}

// -------------------------------- per-token: GAT layer1 (factorized) + layer2
#define X1S 257
__global__ void k_token(const float* __restrict__ LO, const float* __restrict__ HT,
                        const float* __restrict__ ah, const int* __restrict__ adj,
                        const float* __restrict__ HINT, const float* __restrict__ F1I,
                        const float* __restrict__ F2I, const float* __restrict__ Z1,
                        const float* __restrict__ V1, const float* __restrict__ WOA,
                        const float* __restrict__ Wo, float* __restrict__ G) {
  extern __shared__ float sm[];
  float* x1   = sm;                 // [128][X1S]
  float* h0s  = x1 + NN_ * X1S;     // [256]
  float* redt = h0s + H_;           // [256]
  float* f10  = redt + H_;          // [4]
  float* f20  = f10 + 4;            // [4]
  float* f2s  = f20 + 4;            // [128]
  float* alph = f2s + NN_;          // [128]
  float* agg  = alph + NN_;         // [256]
  float* wrow = agg + H_;           // [4*128]
  float* dsum = wrow + 4 * NN_;     // [4]
  float* sc   = dsum + 4;           // [4]
  int m = blockIdx.x, b = m >> 6;
  int tid = threadIdx.x;
  __builtin_prefetch(Wo + tid * H_);

  h0s[tid] = HT[m * H_ + tid];
  __syncthreads();
  // f1_0/f2_0 per head via segmented reduction over 64-wide groups
  int kk = tid >> 6, gg = tid & 63;
  redt[tid] = h0s[tid] * ah[kk * 2 * GH_ + gg];
  __syncthreads();
  for (int off = 32; off; off >>= 1) { if (gg < off) redt[tid] += redt[tid + off]; __syncthreads(); }
  if (gg == 0) f10[kk] = redt[tid];
  __syncthreads();
  redt[tid] = h0s[tid] * ah[kk * 2 * GH_ + GH_ + gg];
  __syncthreads();
  for (int off = 32; off; off >>= 1) { if (gg < off) redt[tid] += redt[tid + off]; __syncthreads(); }
  if (gg == 0) f20[kk] = redt[tid];
  __syncthreads();

  // layer-1 rows i>=1, feature c = k*64+g (head-major)
  for (int idx = tid; idx < NN_ * H_; idx += 256) {
    int i = idx >> 8, c = idx & 255;
    if (i >= 1) {
      int k = c >> 6, g = c & 63;
      float w0 = 0.f;
      if (adj[(b * NN_ + i) * NN_ + 0] > 0)
        w0 = expf(leakyf(F1I[k * NN_ + i] + f20[k]));
      float denom = w0 + Z1[(k * B_ + b) * NN_ + i];
      float num = w0 * h0s[c] + V1[((k * B_ + b) * NN_ + i) * GH_ + g];
      x1[i * X1S + c] = eluf(num / denom);
    }
  }
  // layer-1 row 0: per-head full softmax over neighbors
  for (int idx = tid; idx < NH_ * NN_; idx += 256) {
    int k = idx >> 7, j = idx & 127;
    float w = 0.f;
    if (adj[(b * NN_) * NN_ + j] > 0) {
      float f2 = (j == 0) ? f20[k] : F2I[k * NN_ + j];
      w = expf(leakyf(f10[k] + f2));
    }
    wrow[idx] = w;
  }
  __syncthreads();
  if (tid < 4) {
    float s = 0.f;
    for (int j = 0; j < NN_; ++j) s += wrow[tid * NN_ + j];
    dsum[tid] = s;
  }
  __syncthreads();
  {
    int k = tid >> 6, g = tid & 63;
    float acc = wrow[k * NN_] * h0s[tid];
    const float* hk = HINT + k * NN_ * GH_ + g;
    for (int j = 1; j < NN_; ++j) acc += wrow[k * NN_ + j] * hk[j * GH_];
    x1[tid] = eluf(acc / dsum[k]);  // row 0
  }
  __syncthreads();

  // layer-2 (only node 0 needed): folded attention vectors
  if (tid < NN_) {
    const float* xr = x1 + tid * X1S;
    float a2 = 0.f;
    for (int c = 0; c < H_; ++c) a2 += xr[c] * WOA[H_ + c];
    f2s[tid] = a2;
  }
  if (tid == 255) {
    float a1 = 0.f;
    for (int c = 0; c < H_; ++c) a1 += x1[c] * WOA[c];
    sc[0] = a1;  // f1_2 at node 0
  }
  __syncthreads();
  if (tid < NN_) {
    float w = 0.f;
    if (adj[(b * NN_) * NN_ + tid] > 0) w = expf(leakyf(sc[0] + f2s[tid]));
    alph[tid] = w;
  }
  __syncthreads();
  if (tid == 0) {
    float s = 0.f;
    for (int j = 0; j < NN_; ++j) s += alph[j];
    sc[1] = 1.f / s;
  }
  __syncthreads();
  {
    float acc = 0.f;
    for (int j = 0; j < NN_; ++j) acc += alph[j] * x1[j * X1S + tid];
    agg[tid] = acc * sc[1];
  }
  __syncthreads();
  {
    float acc = 0.f;
    for (int c = 0; c < H_; ++c) acc += agg[c] * Wo[c * H_ + tid];
    G[m * H_ + tid] = eluf(acc) + LO[m * H_ + tid];  // residual (node 0 = lo)
  }
}

// ------------------------------------------ logits = G @ Wl + bl, seq masked
__global__ void k_logits(const float* __restrict__ G, const float* __restrict__ Wl,
                         const float* __restrict__ bl, const int* __restrict__ seq_lens,
                         float* __restrict__ OUT) {
  int wave = threadIdx.x >> 5, lane = threadIdx.x & 31;
  int tile = blockIdx.x * 8 + wave;  // 1024 tiles: 128 x 8
  int m0 = (tile >> 3) * 16, n0 = (tile & 7) * 16;
  int row = m0 + (lane & 15), col = n0 + (lane & 15);
  int kh = (lane >> 4) << 1;
  v8f acc = {};
  for (int k0 = 0; k0 < H_; k0 += 4) {
    v2f a, b;
    a.x = G[row * H_ + k0 + kh];
    a.y = G[row * H_ + k0 + kh + 1];
    b.x = Wl[(k0 + kh) * V_ + col];
    b.y = Wl[(k0 + kh + 1) * V_ + col];
    acc = wmma4(a, b, acc);
  }
  int rbase = m0 + ((lane >> 4) << 3);
  float bias = bl[col];
  for (int r = 0; r < 8; ++r) {
    int mm = rbase + r;
    int bb = mm >> 6, ss = mm & 63;
    OUT[mm * V_ + col] = (ss < seq_lens[bb]) ? (acc[r] + bias) : 0.f;
  }
}

extern "C" void kernel_launch(void* const* d_in, const int* in_sizes, int n_in,
                              void* d_out, int out_size, void* d_ws, size_t ws_size,
                              hipStream_t stream) {
  const float* hidden = (const float*)d_in[0];
  const int*   seq_lens = (const int*)d_in[1];
  const int*   forced = (const int*)d_in[2];
  const int*   adj = (const int*)d_in[3];
  const float* intent = (const float*)d_in[4];
  const float* embedding = (const float*)d_in[5];
  const float* init_t = (const float*)d_in[6];
  const float* W_ih = (const float*)d_in[7];
  const float* W_hh = (const float*)d_in[8];
  const float* b_ih = (const float*)d_in[9];
  const float* b_hh = (const float*)d_in[10];
  const float* Wh = (const float*)d_in[11];
  const float* ah = (const float*)d_in[12];
  const float* Wo = (const float*)d_in[13];
  const float* ao = (const float*)d_in[14];
  const float* Wl = (const float*)d_in[15];
  const float* bl = (const float*)d_in[16];
  float* OUT = (float*)d_out;

  float* ws = (float*)d_ws;
  float* X    = ws;                  // 2048*384
  float* WihT = X + M_ * DX_;        // 384*1024
  float* WhhT = WihT + DX_ * G4H_;   // 256*1024
  float* P    = WhhT + H_ * G4H_;    // 2048*1024
  float* LO   = P + M_ * G4H_;       // 2048*256
  float* HINT = LO + M_ * H_;        // 4*128*64
  float* F1I  = HINT + NH_ * NN_ * GH_;  // 4*128
  float* F2I  = F1I + NH_ * NN_;         // 4*128
  float* WOA  = F2I + NH_ * NN_;         // 2*256
  float* HT   = WOA + 2 * H_;            // 2048*256
  float* Z1   = HT + M_ * H_;            // 4*32*128
  float* V1   = Z1 + NH_ * B_ * NN_;     // 4*32*128*64
  float* G    = V1 + NH_ * B_ * NN_ * GH_;  // 2048*256

  k_transpose<<<(G4H_ * DX_ + 255) / 256, 256, 0, stream>>>(W_ih, WihT, G4H_, DX_);
  k_transpose<<<(G4H_ * H_ + 255) / 256, 256, 0, stream>>>(W_hh, WhhT, G4H_, H_);
  k_build_x<<<(M_ * DX_) / 256, 256, 0, stream>>>(hidden, forced, embedding, init_t, X);
  k_hint<<<dim3(NH_, NN_), GH_, 0, stream>>>(intent, Wh, ah, HINT, F1I, F2I);
  k_woa<<<2, H_, 0, stream>>>(Wo, ao, WOA);

  k_gemm_p<<<1024, 256, 0, stream>>>(X, WihT, b_ih, b_hh, P);

  size_t lstm_lds = (size_t)(B_ * HCS + B_ * G4H_) * sizeof(float);
  k_lstm<<<1, 1024, lstm_lds, stream>>>(P, WhhT, LO);

  k_gemm_ht<<<256, 256, 0, stream>>>(LO, Wh, HT);

  size_t g1_lds = (size_t)(NN_ * WMS + NN_ * GH_) * sizeof(float);
  k_g1<<<dim3(NH_, B_), 256, g1_lds, stream>>>(adj, HINT, F1I, F2I, Z1, V1);

  size_t tok_lds = (size_t)(NN_ * X1S + H_ + H_ + 4 + 4 + NN_ + NN_ + H_ +
                            4 * NN_ + 4 + 4) * sizeof(float);
  k_token<<<M_, 256, tok_lds, stream>>>(LO, HT, ah, adj, HINT, F1I, F2I, Z1, V1,
                                        WOA, Wo, G);

  k_logits<<<128, 256, 0, stream>>>(G, Wl, bl, seq_lens, OUT);
}